// PredictionNetworkForVectorInput_7541962572371
// MI455X (gfx1250) — compile-verified
//
#include <hip/hip_runtime.h>
#include <hip/hip_bf16.h>

// Problem dims (from reference setup_inputs)
#define B_  16
#define T_  16
#define L_  128
#define X_  256
#define H_  512
#define Y_  256

// Per-example fast-weight block layout (float offsets)
#define OFF_FC1 0            // H*X = 131072
#define OFF_B1  131072       // H
#define OFF_FC2 131584       // H*H = 262144
#define OFF_B2  393728       // H
#define OFF_FC3 394240       // H*H
#define OFF_B3  656384       // H
#define OFF_FC4 656896       // Y*H = 131072
#define PSTRIDE 787968       // total floats per example

// Workspace layout (float offsets)
#define WS_WBUF  0
#define WS_ACTA  (B_ * PSTRIDE)                 // H1 / HG buffer [B,L,H]
#define WS_ACTB  (WS_ACTA + B_ * L_ * H_)       // H2 buffer [B,L,H]
#define WS_LOGIT (WS_ACTB + B_ * L_ * H_)       // [B,L,Y]

typedef __attribute__((ext_vector_type(2))) float v2f;
typedef __attribute__((ext_vector_type(8))) float v8f;

// ---------------------------------------------------------------------------
// 1) Replicate shared params into B per-example fast-weight blocks.
// ---------------------------------------------------------------------------
__global__ void init_weights(float* __restrict__ wbuf,
                             const float* __restrict__ fc1, const float* __restrict__ b1,
                             const float* __restrict__ fc2, const float* __restrict__ b2,
                             const float* __restrict__ fc3, const float* __restrict__ b3,
                             const float* __restrict__ fc4) {
  long total = (long)B_ * PSTRIDE;
  for (long idx = blockIdx.x * (long)blockDim.x + threadIdx.x; idx < total;
       idx += (long)gridDim.x * blockDim.x) {
    int o = (int)(idx % PSTRIDE);
    float v;
    if      (o < OFF_B1)  v = fc1[o];
    else if (o < OFF_FC2) v = b1[o - OFF_B1];
    else if (o < OFF_B2)  v = fc2[o - OFF_FC2];
    else if (o < OFF_FC3) v = b2[o - OFF_B2];
    else if (o < OFF_B3)  v = fc3[o - OFF_FC3];
    else if (o < OFF_FC4) v = b3[o - OFF_B3];
    else                  v = fc4[o - OFF_FC4];
    wbuf[idx] = v;
  }
}

// ---------------------------------------------------------------------------
// 2) One MAML inner SGD step. Grid = B workgroups, 512 threads each.
//    Sequential fwd -> bwd -> rank-1 updates; activations live in LDS.
// ---------------------------------------------------------------------------
__global__ __launch_bounds__(512)
void inner_step(float* __restrict__ wbuf,
                const float* __restrict__ train_x, const float* __restrict__ train_y,
                const float* __restrict__ train_gate,
                const float* __restrict__ log_lr, int t) {
  __shared__ float sx[X_], sh1[H_], sh2[H_], sh3[H_], shg[H_];
  __shared__ float sd4[Y_], sd3[H_], sd2[H_], sd1[H_];

  const int b   = blockIdx.x;
  const int tid = threadIdx.x;
  const float lr = __expf(log_lr[0]);

  float* W = wbuf + (size_t)b * PSTRIDE;
  const float* x = train_x    + ((size_t)b * T_ + t) * X_;
  const float* y = train_y    + ((size_t)b * T_ + t) * Y_;
  const float* g = train_gate + ((size_t)b * T_ + t) * H_;

  if (tid < X_) sx[tid] = x[tid];
  __syncthreads();

  // ---- forward ----
  { // h1 = relu(W1 x + b1)
    const float* w = W + OFF_FC1 + (size_t)tid * X_;
    float acc = W[OFF_B1 + tid];
    for (int k = 0; k < X_; ++k) acc = fmaf(w[k], sx[k], acc);
    sh1[tid] = fmaxf(acc, 0.f);
  }
  __syncthreads();
  { // h2 = relu(W2 h1 + b2)
    const float* w = W + OFF_FC2 + (size_t)tid * H_;
    float acc = W[OFF_B2 + tid];
    for (int k = 0; k < H_; ++k) acc = fmaf(w[k], sh1[k], acc);
    sh2[tid] = fmaxf(acc, 0.f);
  }
  __syncthreads();
  { // h3 = relu(W3 h2 + b3); hg = h3 * gate
    const float* w = W + OFF_FC3 + (size_t)tid * H_;
    float acc = W[OFF_B3 + tid];
    for (int k = 0; k < H_; ++k) acc = fmaf(w[k], sh2[k], acc);
    float h3 = fmaxf(acc, 0.f);
    sh3[tid] = h3;
    shg[tid] = h3 * g[tid];
  }
  __syncthreads();
  if (tid < Y_) { // logit = W4 hg ; dlogit = 2/Y * (logit - y)
    const float* w = W + OFF_FC4 + (size_t)tid * H_;
    float acc = 0.f;
    for (int k = 0; k < H_; ++k) acc = fmaf(w[k], shg[k], acc);
    sd4[tid] = (2.0f / (float)Y_) * (acc - y[tid]);
  }
  __syncthreads();

  // ---- backward (uses pre-update weights) ----
  { // dz3 = (W4^T d4) * gate * (h3>0)
    const float* w = W + OFF_FC4 + tid;
    float acc = 0.f;
    for (int r = 0; r < Y_; ++r) acc = fmaf(w[(size_t)r * H_], sd4[r], acc);
    sd3[tid] = (sh3[tid] > 0.f) ? acc * g[tid] : 0.f;
  }
  __syncthreads();
  { // dz2 = (W3^T dz3) * (h2>0)
    const float* w = W + OFF_FC3 + tid;
    float acc = 0.f;
    for (int r = 0; r < H_; ++r) acc = fmaf(w[(size_t)r * H_], sd3[r], acc);
    sd2[tid] = (sh2[tid] > 0.f) ? acc : 0.f;
  }
  __syncthreads();
  { // dz1 = (W2^T dz2) * (h1>0)
    const float* w = W + OFF_FC2 + tid;
    float acc = 0.f;
    for (int r = 0; r < H_; ++r) acc = fmaf(w[(size_t)r * H_], sd2[r], acc);
    sd1[tid] = (sh1[tid] > 0.f) ? acc : 0.f;
  }
  __syncthreads();

  // ---- rank-1 SGD updates ----
  for (int i = tid; i < Y_ * H_; i += blockDim.x) {   // W4 -= lr * d4 (x) hg
    int r = i >> 9, k = i & (H_ - 1);
    W[OFF_FC4 + i] -= lr * sd4[r] * shg[k];
  }
  for (int i = tid; i < H_ * H_; i += blockDim.x) {   // W3 -= lr * dz3 (x) h2
    int r = i >> 9, k = i & (H_ - 1);
    W[OFF_FC3 + i] -= lr * sd3[r] * sh2[k];
  }
  for (int i = tid; i < H_ * H_; i += blockDim.x) {   // W2 -= lr * dz2 (x) h1
    int r = i >> 9, k = i & (H_ - 1);
    W[OFF_FC2 + i] -= lr * sd2[r] * sh1[k];
  }
  for (int i = tid; i < H_ * X_; i += blockDim.x) {   // W1 -= lr * dz1 (x) x
    int r = i >> 8, k = i & (X_ - 1);
    W[OFF_FC1 + i] -= lr * sd1[r] * sx[k];
  }
  W[OFF_B3 + tid] -= lr * sd3[tid];
  W[OFF_B2 + tid] -= lr * sd2[tid];
  W[OFF_B1 + tid] -= lr * sd1[tid];
}

// ---------------------------------------------------------------------------
// 3) Test-eval GEMM: Out[l,n] = epilogue( sum_k In[l,k] * W[n,k] + bias[n] )
//    One wave per 16x64 output strip (4 C tiles); fp32 WMMA 16x16x4.
//    A frag (16x4): lanes 0-15 hold K={k0,k0+1}, lanes 16-31 K={k0+2,k0+3};
//    loaded directly as a lane-addressed float2 (no cndmask selects).
//    Each A fragment is reused by 4 WMMAs (4 N-tiles).
//    C frag: row = mt*16 + r + 8*(lane/16), col = ntile*16 + (lane&15).
// ---------------------------------------------------------------------------
__global__ __launch_bounds__(128)
void eval_gemm(const float* __restrict__ inBase, long inStride,
               const float* __restrict__ wbuf, int woff, int boff,
               const float* __restrict__ gate,       // test_gate [B,L,N] or null
               float* __restrict__ outBase, long outStride,
               int K, int Ntiles, int doRelu) {
  const int wave = (blockIdx.x * blockDim.x + threadIdx.x) >> 5;
  const int lane = threadIdx.x & 31;

  const int NT4 = Ntiles >> 2;                       // groups of 4 N tiles
  const int ng = wave % NT4;
  const int mt = (wave / NT4) & 7;                   // L_/16 == 8 M tiles
  const int b  = wave / (NT4 * 8);

  const float* In = inBase + (size_t)b * inStride;
  const float* W  = wbuf + (size_t)b * PSTRIDE + woff;
  const int half  = lane >> 4;                       // which K pair this lane holds
  const int lr16  = lane & 15;
  const int arow  = mt * 16 + lr16;                  // In row (M)
  const int nbase = ng * 64;                         // first output column of strip

  // lane-addressed fragment pointers: half 0 -> K={k0,k0+1}, half 1 -> K={k0+2,k0+3}
  const float* aPtr = In + (size_t)arow * K + 2 * half;
  const float* bPtr = W + (size_t)(nbase + lr16) * K + 2 * half;

  v8f cacc[4] = {v8f{}, v8f{}, v8f{}, v8f{}};
  for (int k0 = 0; k0 < K; k0 += 4) {
    const v2f a  = *(const v2f*)(aPtr + k0);
    const v2f b0 = *(const v2f*)(bPtr + k0);
    const v2f b1 = *(const v2f*)(bPtr + (size_t)16 * K + k0);
    const v2f b2 = *(const v2f*)(bPtr + (size_t)32 * K + k0);
    const v2f b3 = *(const v2f*)(bPtr + (size_t)48 * K + k0);
    cacc[0] = __builtin_amdgcn_wmma_f32_16x16x4_f32(false, a, false, b0,
                                                    (short)0, cacc[0], false, false);
    cacc[1] = __builtin_amdgcn_wmma_f32_16x16x4_f32(false, a, false, b1,
                                                    (short)0, cacc[1], false, false);
    cacc[2] = __builtin_amdgcn_wmma_f32_16x16x4_f32(false, a, false, b2,
                                                    (short)0, cacc[2], false, false);
    cacc[3] = __builtin_amdgcn_wmma_f32_16x16x4_f32(false, a, false, b3,
                                                    (short)0, cacc[3], false, false);
  }

  const int N = Ntiles * 16;
  float* Out = outBase + (size_t)b * outStride;
#pragma unroll
  for (int j = 0; j < 4; ++j) {
    const int col = nbase + j * 16 + lr16;
    const float bias = (boff >= 0) ? wbuf[(size_t)b * PSTRIDE + boff + col] : 0.f;
#pragma unroll
    for (int r = 0; r < 8; ++r) {
      const int row = mt * 16 + 8 * half + r;
      float v = cacc[j][r] + bias;
      if (doRelu) v = fmaxf(v, 0.f);
      if (gate)   v *= gate[((size_t)b * L_ + row) * N + col];
      Out[(size_t)row * N + col] = v;
    }
  }
}

// ---------------------------------------------------------------------------
// 4) Per-row MSE + output assembly.
//    d_out = [ loss(B*L) | inner_lr(1) | evaluation(B*L) | logit_flat(B*L*Y) ]
// ---------------------------------------------------------------------------
__global__ __launch_bounds__(256)
void finalize(const float* __restrict__ logitBuf, const float* __restrict__ test_y,
              const float* __restrict__ log_lr, float* __restrict__ out) {
  const int row = blockIdx.x;          // 0 .. B*L-1
  const int tid = threadIdx.x;         // 256 == Y_
  __shared__ float red[256];

  const float v = logitBuf[(size_t)row * Y_ + tid];
  const float d = v - test_y[(size_t)row * Y_ + tid];
  red[tid] = d * d;
  out[(size_t)(B_ * L_) + 1 + (size_t)(B_ * L_) + (size_t)row * Y_ + tid] = v;
  __syncthreads();
  for (int s = 128; s > 0; s >>= 1) {
    if (tid < s) red[tid] += red[tid + s];
    __syncthreads();
  }
  if (tid == 0) {
    const float loss = red[0] / (float)Y_;
    out[row] = loss;                         // loss [B,L,1]
    out[B_ * L_ + 1 + row] = loss;           // evaluation [(B L)]
    if (row == 0) out[B_ * L_] = __expf(log_lr[0]);  // inner_lr
  }
}

// ---------------------------------------------------------------------------
extern "C" void kernel_launch(void* const* d_in, const int* in_sizes, int n_in,
                              void* d_out, int out_size, void* d_ws, size_t ws_size,
                              hipStream_t stream) {
  const float* train_x    = (const float*)d_in[0];
  const float* train_y    = (const float*)d_in[1];
  const float* test_x     = (const float*)d_in[2];
  const float* test_y     = (const float*)d_in[3];
  const float* train_gate = (const float*)d_in[4];
  const float* test_gate  = (const float*)d_in[5];
  const float* fc1 = (const float*)d_in[6];
  const float* b1  = (const float*)d_in[7];
  const float* fc2 = (const float*)d_in[8];
  const float* b2  = (const float*)d_in[9];
  const float* fc3 = (const float*)d_in[10];
  const float* b3  = (const float*)d_in[11];
  const float* fc4 = (const float*)d_in[12];
  const float* log_lr = (const float*)d_in[13];
  (void)in_sizes; (void)n_in; (void)out_size; (void)ws_size;

  float* ws       = (float*)d_ws;
  float* Wbuf     = ws + WS_WBUF;
  float* actA     = ws + WS_ACTA;
  float* actB     = ws + WS_ACTB;
  float* logitBuf = ws + WS_LOGIT;

  // 1) replicate shared params -> per-example fast weights (L2-resident)
  init_weights<<<4096, 256, 0, stream>>>(Wbuf, fc1, b1, fc2, b2, fc3, b3, fc4);

  // 2) T sequential inner SGD steps, one workgroup per example
  for (int t = 0; t < T_; ++t)
    inner_step<<<B_, 512, 0, stream>>>(Wbuf, train_x, train_y, train_gate, log_lr, t);

  // 3) test evaluation, fp32 WMMA GEMMs
  //    waves = B * 8 Mtiles * (Ntiles/4); 4 waves per 128-thread block
  eval_gemm<<<(B_ * 8 * (32 / 4)) / 4, 128, 0, stream>>>(   // H1 = relu(X W1^T + b1)
      test_x, (long)L_ * X_, Wbuf, OFF_FC1, OFF_B1, nullptr,
      actA, (long)L_ * H_, X_, 32, 1);
  eval_gemm<<<(B_ * 8 * (32 / 4)) / 4, 128, 0, stream>>>(   // H2 = relu(H1 W2^T + b2)
      actA, (long)L_ * H_, Wbuf, OFF_FC2, OFF_B2, nullptr,
      actB, (long)L_ * H_, H_, 32, 1);
  eval_gemm<<<(B_ * 8 * (32 / 4)) / 4, 128, 0, stream>>>(   // HG = relu(H2 W3^T + b3)*gate
      actB, (long)L_ * H_, Wbuf, OFF_FC3, OFF_B3, test_gate,
      actA, (long)L_ * H_, H_, 32, 1);
  eval_gemm<<<(B_ * 8 * (16 / 4)) / 4, 128, 0, stream>>>(   // LOGIT = HG W4^T
      actA, (long)L_ * H_, Wbuf, OFF_FC4, -1, nullptr,
      logitBuf, (long)L_ * Y_, H_, 16, 0);

  // 4) losses + output assembly
  finalize<<<B_ * L_, 256, 0, stream>>>(logitBuf, test_y, log_lr, (float*)d_out);
}